// SparseMultiHeadAttentionConv_59253368816027
// MI455X (gfx1250) — compile-verified
//
#include <hip/hip_runtime.h>
#include <math.h>

#define N_NODES 50000
#define N_EDGES 800000
#define DIM     128
#define HEADS   8
#define HDIM    16
#define EDIM    16

typedef __attribute__((ext_vector_type(2))) float v2f;
typedef __attribute__((ext_vector_type(8))) float v8f;

// ---------------------------------------------------------------------------
// float atomic max via signed/unsigned int trick (works for any sign, with
// buffer initialized to -inf = 0xFF800000)
// ---------------------------------------------------------------------------
__device__ __forceinline__ void atomicMaxF(float* addr, float val) {
  if (val >= 0.0f) {
    atomicMax((int*)addr, __float_as_int(val));
  } else {
    atomicMin((unsigned int*)addr, __float_as_uint(val));
  }
}

// ---------------------------------------------------------------------------
// Kernel 0: initialize scratch (smax = -inf, denom = 0, agg = 0)
// ---------------------------------------------------------------------------
__global__ void init_kernel(float* __restrict__ smax, float* __restrict__ denom,
                            float* __restrict__ agg) {
  int i = blockIdx.x * blockDim.x + threadIdx.x;
  if (i < N_NODES * HEADS) {
    smax[i]  = -INFINITY;
    denom[i] = 0.0f;
  }
  if (i < N_NODES * DIM) {
    agg[i] = 0.0f;
  }
}

// ---------------------------------------------------------------------------
// Kernel 1: fused QKV projection, q/k/v = h @ W{q,k,v} + b  via fp32 WMMA.
// grid.x = N/16 row tiles; 8 waves per block, wave j owns column tile j.
// A-frag layout (16x4 f32): lanes 0-15 hold K={k0,k0+1}, lanes 16-31 K={k0+2,k0+3}.
// C/D layout (16x16 f32): VGPR r, lanes 0-15 -> M=r, lanes 16-31 -> M=r+8.
// ---------------------------------------------------------------------------
__global__ __launch_bounds__(256) void qkv_kernel(
    const float* __restrict__ h,
    const float* __restrict__ Wq, const float* __restrict__ bq,
    const float* __restrict__ Wk, const float* __restrict__ bk,
    const float* __restrict__ Wv, const float* __restrict__ bv,
    float* __restrict__ q, float* __restrict__ k, float* __restrict__ v) {
  const int lane = threadIdx.x & 31;
  const int wave = threadIdx.x >> 5;     // column tile 0..7
  const int m    = lane & 15;            // A row / B col within tile
  const int kh   = lane >> 4;            // K half select
  const int row0 = blockIdx.x * 16;
  const int col  = wave * 16 + m;

  const float* hrow = h + (size_t)(row0 + m) * DIM;

  v8f cq = {}, ck = {}, cv = {};
  for (int k0 = 0; k0 < DIM; k0 += 4) {
    const int ka = k0 + kh * 2;
    v2f a;
    a[0] = hrow[ka];
    a[1] = hrow[ka + 1];
    v2f bqf, bkf, bvf;
    bqf[0] = Wq[(size_t)ka * DIM + col];
    bqf[1] = Wq[(size_t)(ka + 1) * DIM + col];
    bkf[0] = Wk[(size_t)ka * DIM + col];
    bkf[1] = Wk[(size_t)(ka + 1) * DIM + col];
    bvf[0] = Wv[(size_t)ka * DIM + col];
    bvf[1] = Wv[(size_t)(ka + 1) * DIM + col];
    cq = __builtin_amdgcn_wmma_f32_16x16x4_f32(false, a, false, bqf, (short)0, cq, false, false);
    ck = __builtin_amdgcn_wmma_f32_16x16x4_f32(false, a, false, bkf, (short)0, ck, false, false);
    cv = __builtin_amdgcn_wmma_f32_16x16x4_f32(false, a, false, bvf, (short)0, cv, false, false);
  }

  const float biasq = bq[col], biask = bk[col], biasv = bv[col];
#pragma unroll
  for (int r = 0; r < 8; ++r) {
    const int row = row0 + r + kh * 8;
    q[(size_t)row * DIM + col] = cq[r] + biasq;
    k[(size_t)row * DIM + col] = ck[r] + biask;
    v[(size_t)row * DIM + col] = cv[r] + biasv;
  }
}

// ---------------------------------------------------------------------------
// Kernel 2: per-(edge, head) score = 0.25 * <k[src], q[dst]> + e_feat @ We + be
// plus segment max into smax[dst] via atomics. All gathers as float4 (b128).
// ---------------------------------------------------------------------------
__global__ void score_kernel(
    const float* __restrict__ q, const float* __restrict__ k,
    const int* __restrict__ src, const int* __restrict__ dst,
    const float* __restrict__ e_feat,
    const float* __restrict__ We, const float* __restrict__ be,
    float* __restrict__ score, float* __restrict__ smax) {
  const int idx = blockIdx.x * blockDim.x + threadIdx.x;
  if (idx >= N_EDGES * HEADS) return;
  const int e  = idx >> 3;
  const int hh = idx & 7;
  const int s  = src[e];
  const int d  = dst[e];

  const float4* kp = (const float4*)(k + (size_t)s * DIM + hh * HDIM);
  const float4* qp = (const float4*)(q + (size_t)d * DIM + hh * HDIM);
  float acc = 0.0f;
#pragma unroll
  for (int i = 0; i < 4; ++i) {
    const float4 a4 = kp[i];
    const float4 b4 = qp[i];
    acc += a4.x * b4.x + a4.y * b4.y + a4.z * b4.z + a4.w * b4.w;
  }
  acc *= 0.25f;  // HD^-0.5 = 16^-0.5

  const float4* ef = (const float4*)(e_feat + (size_t)e * EDIM);
  float bias = be[hh];
#pragma unroll
  for (int j = 0; j < 4; ++j) {
    const float4 f = ef[j];
    bias += f.x * We[(4 * j + 0) * HEADS + hh];
    bias += f.y * We[(4 * j + 1) * HEADS + hh];
    bias += f.z * We[(4 * j + 2) * HEADS + hh];
    bias += f.w * We[(4 * j + 3) * HEADS + hh];
  }

  const float sc = acc + bias;
  score[idx] = sc;
  atomicMaxF(&smax[d * HEADS + hh], sc);
}

// ---------------------------------------------------------------------------
// Kernel 3: ex = exp(score - smax[dst]); denom[dst] += ex (atomic)
// ---------------------------------------------------------------------------
__global__ void exp_kernel(
    const int* __restrict__ dst, const float* __restrict__ smax,
    float* __restrict__ score, float* __restrict__ denom) {
  const int idx = blockIdx.x * blockDim.x + threadIdx.x;
  if (idx >= N_EDGES * HEADS) return;
  const int e  = idx >> 3;
  const int hh = idx & 7;
  const int d  = dst[e];
  const float ex = expf(score[idx] - smax[d * HEADS + hh]);
  score[idx] = ex;
  atomicAdd(&denom[d * HEADS + hh], ex);
}

// ---------------------------------------------------------------------------
// Kernel 4: agg[dst, hh*16 + 0..15] += v[src, ...] * (ex / denom[dst])
// one thread per (edge, head): 4x float4 gather of v, 16 f32 atomics into
// L2-resident agg. Index/scale loads amortized over 16 channels.
// ---------------------------------------------------------------------------
__global__ void agg_kernel(
    const int* __restrict__ src, const int* __restrict__ dst,
    const float* __restrict__ v, const float* __restrict__ ex,
    const float* __restrict__ denom, float* __restrict__ agg) {
  const int idx = blockIdx.x * blockDim.x + threadIdx.x;
  if (idx >= N_EDGES * HEADS) return;
  const int e  = idx >> 3;
  const int hh = idx & 7;
  const int dn = dst[e];
  const int sn = src[e];
  const float a = ex[idx] / denom[dn * HEADS + hh];

  const float4* vp = (const float4*)(v + (size_t)sn * DIM + hh * HDIM);
  float* ap = agg + (size_t)dn * DIM + hh * HDIM;
#pragma unroll
  for (int i = 0; i < 4; ++i) {
    const float4 t = vp[i];
    atomicAdd(&ap[4 * i + 0], t.x * a);
    atomicAdd(&ap[4 * i + 1], t.y * a);
    atomicAdd(&ap[4 * i + 2], t.z * a);
    atomicAdd(&ap[4 * i + 3], t.w * a);
  }
}

// ---------------------------------------------------------------------------
// Kernel 5: out = agg @ Wo + bo  via fp32 WMMA (same tiling as qkv_kernel)
// ---------------------------------------------------------------------------
__global__ __launch_bounds__(256) void oproj_kernel(
    const float* __restrict__ agg,
    const float* __restrict__ Wo, const float* __restrict__ bo,
    float* __restrict__ out) {
  const int lane = threadIdx.x & 31;
  const int wave = threadIdx.x >> 5;
  const int m    = lane & 15;
  const int kh   = lane >> 4;
  const int row0 = blockIdx.x * 16;
  const int col  = wave * 16 + m;

  const float* arow = agg + (size_t)(row0 + m) * DIM;

  v8f c = {};
  for (int k0 = 0; k0 < DIM; k0 += 4) {
    const int ka = k0 + kh * 2;
    v2f a;
    a[0] = arow[ka];
    a[1] = arow[ka + 1];
    v2f b;
    b[0] = Wo[(size_t)ka * DIM + col];
    b[1] = Wo[(size_t)(ka + 1) * DIM + col];
    c = __builtin_amdgcn_wmma_f32_16x16x4_f32(false, a, false, b, (short)0, c, false, false);
  }

  const float bias = bo[col];
#pragma unroll
  for (int r = 0; r < 8; ++r) {
    const int row = row0 + r + kh * 8;
    out[(size_t)row * DIM + col] = c[r] + bias;
  }
}

// ---------------------------------------------------------------------------
// Host launch
// ---------------------------------------------------------------------------
extern "C" void kernel_launch(void* const* d_in, const int* in_sizes, int n_in,
                              void* d_out, int out_size, void* d_ws, size_t ws_size,
                              hipStream_t stream) {
  const float* h      = (const float*)d_in[0];
  const int*   src    = (const int*)d_in[1];
  const int*   dst    = (const int*)d_in[2];
  const float* e_feat = (const float*)d_in[3];
  const float* Wq     = (const float*)d_in[4];
  const float* bq     = (const float*)d_in[5];
  const float* Wk     = (const float*)d_in[6];
  const float* bk     = (const float*)d_in[7];
  const float* Wv     = (const float*)d_in[8];
  const float* bv     = (const float*)d_in[9];
  const float* We     = (const float*)d_in[10];
  const float* be     = (const float*)d_in[11];
  const float* Wo     = (const float*)d_in[12];
  const float* bo     = (const float*)d_in[13];
  float* out = (float*)d_out;

  // workspace layout
  char* ws = (char*)d_ws;
  const size_t nd_bytes = (size_t)N_NODES * DIM * sizeof(float);     // 25.6 MB
  const size_t eh_bytes = (size_t)N_EDGES * HEADS * sizeof(float);   // 25.6 MB
  const size_t nh_bytes = (size_t)N_NODES * HEADS * sizeof(float);   // 1.6 MB
  float* q     = (float*)ws;               ws += nd_bytes;
  float* k     = (float*)ws;               ws += nd_bytes;
  float* v     = (float*)ws;               ws += nd_bytes;
  float* ex    = (float*)ws;               ws += eh_bytes;  // scores, then exp
  float* smax  = (float*)ws;               ws += nh_bytes;
  float* denom = (float*)ws;               ws += nh_bytes;
  float* agg   = (float*)ws;               ws += nd_bytes;

  const int threads = 256;

  // 0) init scratch
  {
    int total = N_NODES * DIM;  // covers N*HEADS too
    init_kernel<<<(total + threads - 1) / threads, threads, 0, stream>>>(smax, denom, agg);
  }
  // 1) fused QKV projection (WMMA)
  qkv_kernel<<<N_NODES / 16, threads, 0, stream>>>(h, Wq, bq, Wk, bk, Wv, bv, q, k, v);
  // 2) edge scores + segment max
  {
    int total = N_EDGES * HEADS;
    score_kernel<<<(total + threads - 1) / threads, threads, 0, stream>>>(
        q, k, src, dst, e_feat, We, be, ex, smax);
  }
  // 3) exp + segment sum
  {
    int total = N_EDGES * HEADS;
    exp_kernel<<<(total + threads - 1) / threads, threads, 0, stream>>>(dst, smax, ex, denom);
  }
  // 4) weighted scatter aggregation (vectorized, per (edge, head))
  {
    int total = N_EDGES * HEADS;
    agg_kernel<<<(total + threads - 1) / threads, threads, 0, stream>>>(
        src, dst, v, ex, denom, agg);
  }
  // 5) output projection (WMMA)
  oproj_kernel<<<N_NODES / 16, threads, 0, stream>>>(agg, Wo, bo, out);
}